// Model_18262200942851
// MI455X (gfx1250) — compile-verified
//
#include <hip/hip_runtime.h>
#include <math.h>

typedef __attribute__((ext_vector_type(2))) float v2f;
typedef __attribute__((ext_vector_type(8))) float v8f;

#define NB 8
#define NN 512
#define ND 64

__device__ __forceinline__ float red16(float v) {
  // sum across the 16 lanes of each half-wave (xor masks stay inside the half)
  v += __shfl_xor(v, 1, 32);
  v += __shfl_xor(v, 2, 32);
  v += __shfl_xor(v, 4, 32);
  v += __shfl_xor(v, 8, 32);
  return v;
}

__global__ __launch_bounds__(256) void gat_row_kernel(
    const float* __restrict__ x,
    const float* __restrict__ att_proj_w,
    const float* __restrict__ att_proj_b,
    const float* __restrict__ att_weight,
    const float* __restrict__ pwa_w,
    const float* __restrict__ pwa_b,
    const float* __restrict__ pwo_w,
    const float* __restrict__ pwo_b,
    const float* __restrict__ bn_gamma,
    const float* __restrict__ bn_beta,
    const float* __restrict__ pool_w,
    const float* __restrict__ pool_b,
    float* __restrict__ hs_out,   // (B,N,D) h*score
    float* __restrict__ sc_out)   // (B,N)   score
{
  __shared__ float sW[ND * ND];
  __shared__ float sxi[ND];
  __shared__ float sbias[ND];
  __shared__ float saw[ND];
  __shared__ float sLog[NN];
  __shared__ float sRed[256];
  __shared__ float sAgg[ND];
  __shared__ float sH[ND];
  __shared__ float sScore;

  const int t    = threadIdx.x;
  const int lane = t & 31;
  const int wave = t >> 5;
  const int blk  = blockIdx.x;
  const int i    = blk & (NN - 1);
  const int bb   = blk >> 9;
  const float* xb = x + bb * NN * ND;

  // --- stage weights / per-row vector into LDS ---
  for (int idx = t; idx < ND * ND; idx += 256) sW[idx] = att_proj_w[idx];
  if (t < ND) {
    sxi[t]   = xb[i * ND + t];
    sbias[t] = att_proj_b[t];
    saw[t]   = att_weight[t];
  }
  __syncthreads();

  // --- logits[j] = aw . tanh( (x_i (.) x_j) @ W + bias ), via f32 WMMA ---
  const int m    = lane & 15;   // row within tile (lanes 0-15 / 16-31 hold K-halves)
  const int half = lane >> 4;
  for (int jt = wave; jt < NN / 16; jt += 8) {
    v8f acc0 = {}, acc1 = {}, acc2 = {}, acc3 = {};
    const int j = jt * 16 + m;
    for (int d0 = 0; d0 < ND; d0 += 4) {
      const int dd = d0 + 2 * half;
      v2f a;
      a.x = xb[j * ND + dd]     * sxi[dd];
      a.y = xb[j * ND + dd + 1] * sxi[dd + 1];
      v2f w0, w1, w2, w3;
      w0.x = sW[dd * ND + m];            w0.y = sW[(dd + 1) * ND + m];
      w1.x = sW[dd * ND + 16 + m];       w1.y = sW[(dd + 1) * ND + 16 + m];
      w2.x = sW[dd * ND + 32 + m];       w2.y = sW[(dd + 1) * ND + 32 + m];
      w3.x = sW[dd * ND + 48 + m];       w3.y = sW[(dd + 1) * ND + 48 + m];
      acc0 = __builtin_amdgcn_wmma_f32_16x16x4_f32(false, a, false, w0, (short)0, acc0, false, false);
      acc1 = __builtin_amdgcn_wmma_f32_16x16x4_f32(false, a, false, w1, (short)0, acc1, false, false);
      acc2 = __builtin_amdgcn_wmma_f32_16x16x4_f32(false, a, false, w2, (short)0, acc2, false, false);
      acc3 = __builtin_amdgcn_wmma_f32_16x16x4_f32(false, a, false, w3, (short)0, acc3, false, false);
    }
    // tanh + dot with att_weight; column k = kt*16 + (lane&15)
    #pragma unroll
    for (int r = 0; r < 8; ++r) {
      float p = tanhf(acc0[r] + sbias[m])      * saw[m]
              + tanhf(acc1[r] + sbias[16 + m]) * saw[16 + m]
              + tanhf(acc2[r] + sbias[32 + m]) * saw[32 + m]
              + tanhf(acc3[r] + sbias[48 + m]) * saw[48 + m];
      p = red16(p);
      if (m == 0) sLog[jt * 16 + half * 8 + r] = p;   // row = r (+8 for upper half)
    }
  }
  __syncthreads();

  // --- softmax over j (512) ---
  float l0 = sLog[t], l1 = sLog[t + 256];
  sRed[t] = fmaxf(l0, l1);
  __syncthreads();
  for (int s = 128; s > 0; s >>= 1) {
    if (t < s) sRed[t] = fmaxf(sRed[t], sRed[t + s]);
    __syncthreads();
  }
  const float mx = sRed[0];
  __syncthreads();
  float e0 = expf(l0 - mx), e1 = expf(l1 - mx);
  sRed[t] = e0 + e1;
  __syncthreads();
  for (int s = 128; s > 0; s >>= 1) {
    if (t < s) sRed[t] += sRed[t + s];
    __syncthreads();
  }
  const float inv = 1.0f / sRed[0];
  __syncthreads();
  sLog[t] = e0 * inv;
  sLog[t + 256] = e1 * inv;
  __syncthreads();

  // --- agg[d] = sum_j att[j] * x[b,j,d] ---
  {
    const int d  = t & 63;
    const int ch = t >> 6;            // 4 chunks of 128 j
    float p = 0.0f;
    const int j0 = ch * 128;
    for (int j = j0; j < j0 + 128; ++j) p += sLog[j] * xb[j * ND + d];
    sRed[t] = p;
  }
  __syncthreads();
  if (t < 64) sAgg[t] = sRed[t] + sRed[t + 64] + sRed[t + 128] + sRed[t + 192];
  __syncthreads();

  // --- h = agg@pwa_w + x_i@pwo_w + biases; BN(eval); SELU ---
  if (t < 64) {
    float acc = pwa_b[t] + pwo_b[t];
    for (int d = 0; d < ND; ++d)
      acc += sAgg[d] * pwa_w[d * ND + t] + sxi[d] * pwo_w[d * ND + t];
    float hn = acc * rsqrtf(1.0f + 1e-5f) * bn_gamma[t] + bn_beta[t];
    const float SC = 1.0507009873554804934193349852946f;
    const float AL = 1.6732632423543772848170429916717f;
    sH[t] = hn > 0.0f ? SC * hn : SC * AL * expm1f(hn);
  }
  __syncthreads();

  // --- pool score = sigmoid(h . pool_w + pool_b) ---
  if (t == 0) {
    float s = pool_b[0];
    for (int d = 0; d < ND; ++d) s += sH[d] * pool_w[d];
    sScore = 1.0f / (1.0f + expf(-s));
  }
  __syncthreads();

  if (t < 64) hs_out[(bb * NN + i) * ND + t] = sH[t] * sScore;
  if (t == 0) sc_out[bb * NN + i] = sScore;
}

// Exact jax.lax.top_k semantics: descending order, ties -> smaller index first.
__global__ __launch_bounds__(512) void topk_gather_kernel(
    const float* __restrict__ hs, const float* __restrict__ sc,
    const int* __restrict__ kptr, float* __restrict__ out)
{
  __shared__ float s[NN];
  const int bb = blockIdx.x;
  const int t  = threadIdx.x;
  s[t] = sc[bb * NN + t];
  __syncthreads();
  const int K = kptr[0];
  const float my = s[t];
  int rank = 0;
  for (int j = 0; j < NN; ++j) {
    const float vj = s[j];
    rank += ((vj > my) || (vj == my && j < t)) ? 1 : 0;
  }
  if (rank < K) {
    const float* src = hs + (bb * NN + t) * ND;
    float* dst = out + (bb * K + rank) * ND;
    for (int c = 0; c < ND; ++c) dst[c] = src[c];
  }
}

extern "C" void kernel_launch(void* const* d_in, const int* in_sizes, int n_in,
                              void* d_out, int out_size, void* d_ws, size_t ws_size,
                              hipStream_t stream) {
  const float* x          = (const float*)d_in[0];
  const float* att_proj_w = (const float*)d_in[1];
  const float* att_proj_b = (const float*)d_in[2];
  const float* att_weight = (const float*)d_in[3];
  const float* pwa_w      = (const float*)d_in[4];
  const float* pwa_b      = (const float*)d_in[5];
  const float* pwo_w      = (const float*)d_in[6];
  const float* pwo_b      = (const float*)d_in[7];
  const float* bn_gamma   = (const float*)d_in[8];
  const float* bn_beta    = (const float*)d_in[9];
  const float* pool_w     = (const float*)d_in[10];
  const float* pool_b     = (const float*)d_in[11];
  const int*   k_nodes    = (const int*)d_in[12];

  float* ws  = (float*)d_ws;
  float* hs  = ws;                      // B*N*D floats
  float* scw = ws + NB * NN * ND;       // B*N floats

  gat_row_kernel<<<NB * NN, 256, 0, stream>>>(
      x, att_proj_w, att_proj_b, att_weight, pwa_w, pwa_b, pwo_w, pwo_b,
      bn_gamma, bn_beta, pool_w, pool_b, hs, scw);

  topk_gather_kernel<<<NB, NN, 0, stream>>>(hs, scw, k_nodes, (float*)d_out);
}